// SuperLinear_55611236548811
// MI455X (gfx1250) — compile-verified
//
#include <hip/hip_runtime.h>
#include <hip/hip_bf16.h>

// Problem constants (match reference)
#define BB        128
#define NN        4096
#define IN_DIMS   128
#define OUT_DIMS  64
#define LN_EPS    1e-5f
#define WS_STRIDE 132          // padded LDS row stride (floats) -> conflict-free b64 reads

typedef __attribute__((ext_vector_type(2))) float v2f;
typedef __attribute__((ext_vector_type(8))) float v8f;

// ---------------------------------------------------------------------------
// Kernel 1: w_eff[i*64+o] = sum_g w1[i][o][g]   (w1 flat: ((i*64+o)*4096 + g))
// One wave per (i,o) pair: 8192 waves = 1024 blocks x 256 threads.
// ---------------------------------------------------------------------------
__global__ __launch_bounds__(256) void wsum_kernel(const float* __restrict__ w1,
                                                   float* __restrict__ w_eff) {
    const int wid  = (blockIdx.x * blockDim.x + threadIdx.x) >> 5; // 0..8191
    const int lane = threadIdx.x & 31;
    const float4* p = reinterpret_cast<const float4*>(w1 + (long long)wid * NN);
    float s = 0.f;
#pragma unroll 8
    for (int it = 0; it < 32; ++it) {               // 32 lanes * float4 * 32 = 4096
        float4 v = p[it * 32 + lane];
        s += (v.x + v.y) + (v.z + v.w);
    }
#pragma unroll
    for (int m = 16; m >= 1; m >>= 1) s += __shfl_xor(s, m, 32);
    if (lane == 0) w_eff[wid] = s;
}

// ---------------------------------------------------------------------------
// Kernel 2: fused LayerNorm + (xn @ w_eff)*T + b1*T via V_WMMA_F32_16X16X4_F32
// Block = 256 threads = 8 waves; wave handles a 16-row x 64-col output tile.
// ---------------------------------------------------------------------------
__global__ __launch_bounds__(256) void ln_gemm_kernel(const float* __restrict__ x,
                                                      const float* __restrict__ gamma,
                                                      const float* __restrict__ beta,
                                                      const float* __restrict__ b1,
                                                      const float* __restrict__ Tp,
                                                      const float* __restrict__ w_eff,
                                                      float* __restrict__ out) {
    __shared__ float ws[OUT_DIMS * WS_STRIDE];   // w_eff^T * T, padded: ws[o*132 + k]
    __shared__ float gs[IN_DIMS];
    __shared__ float bs[IN_DIMS];

    const int tid  = threadIdx.x;
    const int wave = tid >> 5;
    const int lane = tid & 31;
    const int r    = lane & 15;   // row within tile (A) / col within 16-col tile (B,D)
    const int h    = lane >> 4;   // half-wave select
    const float T  = Tp[0];

    // ---- cooperative staging: gamma/beta + transposed, T-scaled w_eff ----
    if (tid < IN_DIMS) { gs[tid] = gamma[tid]; bs[tid] = beta[tid]; }
#pragma unroll
    for (int i = tid; i < IN_DIMS * OUT_DIMS; i += 256) {
        const int k = i >> 6;          // i = k*64 + o
        const int o = i & 63;
        ws[o * WS_STRIDE + k] = w_eff[i] * T;
    }

    // ---- load this lane's A-fragment straight into registers ----
    // 32-bit A 16x4 layout: lanes 0-15 hold K = 4j, 4j+1 ; lanes 16-31 hold K = 4j+2, 4j+3
    const long long row_base = (long long)blockIdx.x * 128 + wave * 16;
    const float* xrow = x + (row_base + r) * (long long)IN_DIMS + 2 * h;
    v2f xv[32];
#pragma unroll
    for (int j = 0; j < 32; ++j)
        xv[j] = *reinterpret_cast<const v2f*>(xrow + 4 * j);

    // ---- LayerNorm stats from registers (lane pair L / L+16 covers full row) ----
    float s = 0.f, q = 0.f;
#pragma unroll
    for (int j = 0; j < 32; ++j) {
        s += xv[j][0] + xv[j][1];
        q += xv[j][0] * xv[j][0] + xv[j][1] * xv[j][1];
    }
    s += __shfl_xor(s, 16, 32);
    q += __shfl_xor(q, 16, 32);
    const float mu   = s * (1.0f / IN_DIMS);
    const float var  = fmaf(-mu, mu, q * (1.0f / IN_DIMS));
    const float rstd = rsqrtf(var + LN_EPS);

    __syncthreads();   // ws / gs / bs ready

    // ---- normalize + affine in registers ----
#pragma unroll
    for (int j = 0; j < 32; ++j) {
        const v2f g = *reinterpret_cast<const v2f*>(gs + 4 * j + 2 * h);
        const v2f b = *reinterpret_cast<const v2f*>(bs + 4 * j + 2 * h);
        xv[j][0] = fmaf((xv[j][0] - mu) * rstd, g[0], b[0]);
        xv[j][1] = fmaf((xv[j][1] - mu) * rstd, g[1], b[1]);
    }

    // ---- K-loop: 32 steps of 16x16x4 f32 WMMA, 4 output column tiles ----
    v8f acc[4] = {v8f{}, v8f{}, v8f{}, v8f{}};
    // B 4x16 layout mirrors A: lane (h,r) supplies B[K=4j+2h][o], B[K=4j+2h+1][o], o = c*16+r
    const float* bp = ws + r * WS_STRIDE + 2 * h;
#pragma unroll
    for (int j = 0; j < 32; ++j) {
        const v2f b0 = *reinterpret_cast<const v2f*>(bp + 0 * 16 * WS_STRIDE + 4 * j);
        const v2f b1v = *reinterpret_cast<const v2f*>(bp + 1 * 16 * WS_STRIDE + 4 * j);
        const v2f b2v = *reinterpret_cast<const v2f*>(bp + 2 * 16 * WS_STRIDE + 4 * j);
        const v2f b3v = *reinterpret_cast<const v2f*>(bp + 3 * 16 * WS_STRIDE + 4 * j);
        acc[0] = __builtin_amdgcn_wmma_f32_16x16x4_f32(false, xv[j], false, b0,  (short)0, acc[0], false, false);
        acc[1] = __builtin_amdgcn_wmma_f32_16x16x4_f32(false, xv[j], false, b1v, (short)0, acc[1], false, false);
        acc[2] = __builtin_amdgcn_wmma_f32_16x16x4_f32(false, xv[j], false, b2v, (short)0, acc[2], false, false);
        acc[3] = __builtin_amdgcn_wmma_f32_16x16x4_f32(false, xv[j], false, b3v, (short)0, acc[3], false, false);
    }

    // ---- epilogue: out[row, o] = acc + T*b1[n, o]
    // D layout: lane -> col c*16 + r ; VGPR v -> row v + 8h
    const long long rbase = row_base + 8 * h;
#pragma unroll
    for (int c = 0; c < 4; ++c) {
        const int o = c * 16 + r;
#pragma unroll
        for (int v = 0; v < 8; ++v) {
            const long long rr = rbase + v;
            const int n = (int)(rr & (NN - 1));
            out[rr * OUT_DIMS + o] = acc[c][v] + T * b1[(long long)n * OUT_DIMS + o];
        }
    }
}

// ---------------------------------------------------------------------------
extern "C" void kernel_launch(void* const* d_in, const int* in_sizes, int n_in,
                              void* d_out, int out_size, void* d_ws, size_t ws_size,
                              hipStream_t stream) {
    const float* x     = (const float*)d_in[0];
    const float* gamma = (const float*)d_in[1];
    const float* beta  = (const float*)d_in[2];
    const float* w1    = (const float*)d_in[3];
    const float* b1    = (const float*)d_in[4];
    const float* T     = (const float*)d_in[5];
    float* out   = (float*)d_out;
    float* w_eff = (float*)d_ws;     // 128*64 floats = 32 KB scratch

    // Phase 1: reduce w1 over its N axis. 8192 (i,o) pairs, 1 wave each.
    wsum_kernel<<<(IN_DIMS * OUT_DIMS) / 8, 256, 0, stream>>>(w1, w_eff);

    // Phase 2: fused LayerNorm + GEMM + bias + temperature.
    // 524288 rows / (8 waves * 16 rows) = 4096 blocks.
    ln_gemm_kernel<<<(BB * NN) / 128, 256, 0, stream>>>(x, gamma, beta, b1, T, w_eff, out);
}